// ProteinGCN_40518721470743
// MI455X (gfx1250) — compile-verified
//
#include <hip/hip_runtime.h>
#include <hip/hip_bf16.h>
#include <stdint.h>

#define N_NODES 50000
#define N_EDGES 800000
#define N_FEAT 128
#define HIDDEN 128
#define N_CLASSES 10
#define N_GRAPHS 256

typedef __attribute__((ext_vector_type(2))) float v2f;
typedef __attribute__((ext_vector_type(8))) float v8f;

// ---------------------------------------------------------------------------
// Degree / normalization (computed once; depends only on dst)
// ---------------------------------------------------------------------------
__global__ void deg_init(float* __restrict__ deg, int n) {
  int i = blockIdx.x * blockDim.x + threadIdx.x;
  if (i < n) deg[i] = 1.0f;  // self-loop contribution
}

__global__ void deg_accum(const long long* __restrict__ dst, float* __restrict__ deg, int e) {
  int i = blockIdx.x * blockDim.x + threadIdx.x;
  if (i < e) atomicAdd(&deg[(int)dst[i]], 1.0f);
}

__global__ void deg_to_dinv(float* __restrict__ deg, int n) {
  int i = blockIdx.x * blockDim.x + threadIdx.x;
  if (i < n) {
    float d = deg[i];
    deg[i] = (d > 0.0f) ? rsqrtf(d) : 0.0f;  // in-place: deg -> deg^{-1/2}
  }
}

// ---------------------------------------------------------------------------
// Transpose 128x128 weight: Wt[o][k] = W[k][o]. Makes the WMMA B-fragment
// (consecutive k, fixed o) a contiguous 8-byte load in the hot loop.
// ---------------------------------------------------------------------------
__global__ void transpose_w(const float* __restrict__ W, float* __restrict__ Wt) {
  int t = blockIdx.x * blockDim.x + threadIdx.x;  // 16384 threads
  if (t >= HIDDEN * HIDDEN) return;
  int r = t >> 7;        // k
  int c = t & 127;       // o
  Wt[(size_t)c * HIDDEN + r] = W[(size_t)r * HIDDEN + c];
}

// ---------------------------------------------------------------------------
// WMMA GEMM: H[N,128] = act(X[N,128]) @ W[128,128]  (W given transposed)
// One block = 8 waves = 16-row stripe x full 128 cols. 3125 blocks (50000/16).
// A tile (16x128 = 8KB) staged once per block in LDS (ReLU fused at stage);
// B fragments are contiguous b64 loads from the transposed weights.
// Exact-f32 V_WMMA_F32_16X16X4_F32, K stepped by 4 -> 32 WMMAs per tile.
// ---------------------------------------------------------------------------
template <bool RELU_IN>
__global__ __launch_bounds__(256) void gemm_wmma_f32(const float* __restrict__ X,
                                                     const float* __restrict__ Wt,
                                                     float* __restrict__ H) {
  __shared__ float xs[16 * N_FEAT];  // 8 KB

  const int tid  = threadIdx.x;
  const int lane = tid & 31;
  const int wave = tid >> 5;         // 0..7 -> column tile
  const int col0 = wave * 16;
  const int row0 = blockIdx.x * 16;
  const int m    = lane & 15;        // A: M index / B: N index
  const int grp  = lane >> 4;        // K-pair group (0 or 1)

  // Stage 16 rows x 128 feats, coalesced float4 (2048 floats / 256 threads = 2 ea).
  {
    const float4* gsrc = (const float4*)(X + (size_t)row0 * N_FEAT);
    float4* lds = (float4*)xs;
#pragma unroll
    for (int j = 0; j < 2; ++j) {
      int idx = tid + j * 256;       // 0..511
      float4 v = gsrc[idx];
      if (RELU_IN) {
        v.x = fmaxf(v.x, 0.0f);
        v.y = fmaxf(v.y, 0.0f);
        v.z = fmaxf(v.z, 0.0f);
        v.w = fmaxf(v.w, 0.0f);
      }
      lds[idx] = v;
    }
  }
  __syncthreads();

  const float* __restrict__ xrow = xs + m * N_FEAT;               // LDS
  const float* __restrict__ wcol = Wt + (size_t)(col0 + m) * HIDDEN;  // transposed W row
  v8f acc = {};

#pragma unroll
  for (int kb = 0; kb < N_FEAT; kb += 4) {
    const int k = kb + 2 * grp;
    v2f a = *(const v2f*)(xrow + k);   // ds_load_b64
    v2f b = *(const v2f*)(wcol + k);   // global_load_b64 (contiguous)
    // 8 args: (neg_a, A, neg_b, B, c_mod, C, reuse_a, reuse_b)
    acc = __builtin_amdgcn_wmma_f32_16x16x4_f32(false, a, false, b, (short)0, acc,
                                                false, false);
  }

#pragma unroll
  for (int r = 0; r < 8; ++r) {
    const int orow = row0 + r + 8 * grp;  // C/D layout: M = r + 8*(lane>>4)
    H[(size_t)orow * HIDDEN + col0 + m] = acc[r];
  }
}

// ---------------------------------------------------------------------------
// Self-loop + bias init: out[n,:] = H[n,:] * dinv[n]^2 + b
// ---------------------------------------------------------------------------
__global__ __launch_bounds__(256) void selfloop_bias(const float* __restrict__ H,
                                                     const float* __restrict__ dinv,
                                                     const float* __restrict__ bias,
                                                     float* __restrict__ out) {
  int t = blockIdx.x * blockDim.x + threadIdx.x;
  int n = t >> 5;
  if (n >= N_NODES) return;
  int c = (t & 31) * 4;
  float w = dinv[n];
  w = w * w;
  float4 v = *(const float4*)(H + (size_t)n * HIDDEN + c);
  float4 bb = *(const float4*)(bias + c);
  float4 r;
  r.x = v.x * w + bb.x;
  r.y = v.y * w + bb.y;
  r.z = v.z * w + bb.z;
  r.w = v.w * w + bb.w;
  *(float4*)(out + (size_t)n * HIDDEN + c) = r;
}

// ---------------------------------------------------------------------------
// Edge scatter: out[dst,:] += H[src,:] * dinv[src]*dinv[dst]
// One wave per edge; gathers are b128, atomics land in 192MB L2
// (each feature matrix is 25.6MB -> fully L2-resident).
// ---------------------------------------------------------------------------
__global__ __launch_bounds__(256) void edge_scatter(const float* __restrict__ H,
                                                    const float* __restrict__ dinv,
                                                    const long long* __restrict__ src,
                                                    const long long* __restrict__ dst,
                                                    float* __restrict__ out) {
  int e = blockIdx.x * 8 + (threadIdx.x >> 5);
  if (e >= N_EDGES) return;
  int lane = threadIdx.x & 31;
  int s = (int)src[e];
  int d = (int)dst[e];
  float norm = dinv[s] * dinv[d];
  float4 v = *(const float4*)(H + (size_t)s * HIDDEN + lane * 4);
  float* o = out + (size_t)d * HIDDEN + lane * 4;
  atomicAdd(o + 0, v.x * norm);
  atomicAdd(o + 1, v.y * norm);
  atomicAdd(o + 2, v.z * norm);
  atomicAdd(o + 3, v.w * norm);
}

// ---------------------------------------------------------------------------
// Global mean pool + classifier head
// ---------------------------------------------------------------------------
__global__ void pool_zero(float* __restrict__ gsum, float* __restrict__ gcnt) {
  int i = blockIdx.x * blockDim.x + threadIdx.x;
  if (i < N_GRAPHS * HIDDEN) gsum[i] = 0.0f;
  if (i < N_GRAPHS) gcnt[i] = 0.0f;
}

__global__ __launch_bounds__(256) void pool_accum(const float* __restrict__ feat,
                                                  const long long* __restrict__ batch,
                                                  float* __restrict__ gsum,
                                                  float* __restrict__ gcnt) {
  int t = blockIdx.x * blockDim.x + threadIdx.x;
  int n = t >> 5;
  if (n >= N_NODES) return;
  int lane = t & 31;
  int g = (int)batch[n];
  int c = lane * 4;
  float4 v = *(const float4*)(feat + (size_t)n * HIDDEN + c);
  float* o = gsum + (size_t)g * HIDDEN + c;
  atomicAdd(o + 0, v.x);
  atomicAdd(o + 1, v.y);
  atomicAdd(o + 2, v.z);
  atomicAdd(o + 3, v.w);
  if (lane == 0) atomicAdd(&gcnt[g], 1.0f);
}

__global__ void final_linear(const float* __restrict__ gsum, const float* __restrict__ gcnt,
                             const float* __restrict__ Wl, const float* __restrict__ bl,
                             float* __restrict__ out) {
  int t = blockIdx.x * blockDim.x + threadIdx.x;
  if (t >= N_GRAPHS * N_CLASSES) return;
  int g = t / N_CLASSES;
  int c = t % N_CLASSES;
  float inv = 1.0f / fmaxf(gcnt[g], 1.0f);
  float acc = bl[c];
#pragma unroll 8
  for (int f = 0; f < HIDDEN; ++f) acc += gsum[(size_t)g * HIDDEN + f] * inv * Wl[f * N_CLASSES + c];
  out[t] = acc;
}

// ---------------------------------------------------------------------------
// Host-side orchestration
// ---------------------------------------------------------------------------
extern "C" void kernel_launch(void* const* d_in, const int* in_sizes, int n_in,
                              void* d_out, int out_size, void* d_ws, size_t ws_size,
                              hipStream_t stream) {
  (void)in_sizes; (void)n_in; (void)out_size; (void)ws_size;

  const float* x = (const float*)d_in[0];
  const long long* edge = (const long long*)d_in[1];   // [2, E] int64
  const long long* batch = (const long long*)d_in[2];  // [N] int64
  const float* W1 = (const float*)d_in[3];
  const float* b1 = (const float*)d_in[4];
  const float* W2 = (const float*)d_in[5];
  const float* b2 = (const float*)d_in[6];
  const float* W3 = (const float*)d_in[7];
  const float* b3 = (const float*)d_in[8];
  const float* Wl = (const float*)d_in[9];
  const float* bl = (const float*)d_in[10];
  float* out = (float*)d_out;

  const long long* src = edge;            // row 0
  const long long* dst = edge + N_EDGES;  // row 1

  // Workspace partition (floats)
  float* ws = (float*)d_ws;
  float* dinv = ws;                                  // 50000 (rounded to 50176)
  float* tmpH = ws + 50176;                          // 6.4M floats
  float* feat = tmpH + (size_t)N_NODES * HIDDEN;     // 6.4M floats
  float* gsum = feat + (size_t)N_NODES * HIDDEN;     // 32768
  float* gcnt = gsum + N_GRAPHS * HIDDEN;            // 256
  float* Wt   = gcnt + 256;                          // 16384 (reused per layer)

  const int T = 256;
  const int gemm_blocks = N_NODES / 16;              // 3125 (exact)
  const int vec_blocks = (N_NODES * 32 + T - 1) / T;
  const int edge_blocks = (N_EDGES + 7) / 8;         // 8 edges per 256-thread block
  const int wt_blocks = (HIDDEN * HIDDEN + T - 1) / T;

  // --- degree / normalization (once) ---
  deg_init<<<(N_NODES + T - 1) / T, T, 0, stream>>>(dinv, N_NODES);
  deg_accum<<<(N_EDGES + T - 1) / T, T, 0, stream>>>(dst, dinv, N_EDGES);
  deg_to_dinv<<<(N_NODES + T - 1) / T, T, 0, stream>>>(dinv, N_NODES);

  // --- layer 1 ---
  transpose_w<<<wt_blocks, T, 0, stream>>>(W1, Wt);
  gemm_wmma_f32<false><<<gemm_blocks, T, 0, stream>>>(x, Wt, tmpH);
  selfloop_bias<<<vec_blocks, T, 0, stream>>>(tmpH, dinv, b1, feat);
  edge_scatter<<<edge_blocks, T, 0, stream>>>(tmpH, dinv, src, dst, feat);

  // --- layer 2 (ReLU fused into LDS staging) ---
  transpose_w<<<wt_blocks, T, 0, stream>>>(W2, Wt);
  gemm_wmma_f32<true><<<gemm_blocks, T, 0, stream>>>(feat, Wt, tmpH);
  selfloop_bias<<<vec_blocks, T, 0, stream>>>(tmpH, dinv, b2, feat);
  edge_scatter<<<edge_blocks, T, 0, stream>>>(tmpH, dinv, src, dst, feat);

  // --- layer 3 (no output ReLU) ---
  transpose_w<<<wt_blocks, T, 0, stream>>>(W3, Wt);
  gemm_wmma_f32<true><<<gemm_blocks, T, 0, stream>>>(feat, Wt, tmpH);
  selfloop_bias<<<vec_blocks, T, 0, stream>>>(tmpH, dinv, b3, feat);
  edge_scatter<<<edge_blocks, T, 0, stream>>>(tmpH, dinv, src, dst, feat);

  // --- global mean pool + head ---
  pool_zero<<<(N_GRAPHS * HIDDEN + T - 1) / T, T, 0, stream>>>(gsum, gcnt);
  pool_accum<<<vec_blocks, T, 0, stream>>>(feat, batch, gsum, gcnt);
  final_linear<<<(N_GRAPHS * N_CLASSES + T - 1) / T, T, 0, stream>>>(gsum, gcnt, Wl, bl, out);
}